// AttentionSubModule_32212254720703
// MI455X (gfx1250) — compile-verified
//
#include <hip/hip_runtime.h>

typedef __attribute__((ext_vector_type(2))) float v2f;
typedef __attribute__((ext_vector_type(8))) float v8f;
typedef __attribute__((ext_vector_type(4))) float f4;
typedef __attribute__((ext_vector_type(4))) int   v4i;

#define KVD 9
#define NQ  25
#define XW  225          // NQ*KVD floats per batch item
#define TB  16           // batch items per workgroup (M dimension of WMMA)
#define THREADS 128
#define NWAVES 4

#define AS1 __attribute__((address_space(1)))
#define AS3 __attribute__((address_space(3)))

__device__ __forceinline__ void wait_asynccnt0() {
#if __has_builtin(__builtin_amdgcn_s_wait_asynccnt)
    __builtin_amdgcn_s_wait_asynccnt(0);
#else
    asm volatile("s_wait_asynccnt 0x0" ::: "memory");
#endif
}

__launch_bounds__(THREADS)
__global__ void attn_small_kernel(
    const float* __restrict__ x,
    const float* __restrict__ mask,
    const float* __restrict__ Wq, const float* __restrict__ bq,
    const float* __restrict__ Wk, const float* __restrict__ bk,
    const float* __restrict__ Wv, const float* __restrict__ bv,
    const float* __restrict__ gamma, const float* __restrict__ beta,
    float* __restrict__ out, int B)
{
    __shared__ float xs[TB * XW];           // x tile; reused as output staging
    __shared__ float qs [NQ * TB * KVD];
    __shared__ float ks_[NQ * TB * KVD];
    __shared__ float vs [NQ * TB * KVD];
    __shared__ float wts[3 * 4 * 81];       // Wq | Wk | Wv
    __shared__ float bis[3 * 4 * KVD];      // bq | bk | bv
    __shared__ float msk[NQ];               // mask * -1e9 (pre-multiplied)
    __shared__ float gam[KVD], bet[KVD];

    const int tid  = threadIdx.x;
    const int wave = tid >> 5;
    const int lane = tid & 31;
    const int b0   = blockIdx.x * TB;
    const int rows = (B - b0) < TB ? (B - b0) : TB;

    // ---- stage x tile (16*225 contiguous floats), async global->LDS if available ----
    if (rows == TB) {
        const f4* xsrc = (const f4*)(x + (size_t)b0 * XW);
        f4* xdst = (f4*)xs;
#if __has_builtin(__builtin_amdgcn_global_load_async_to_lds_b128)
        for (int t = tid; t < (TB * XW) / 4; t += THREADS)
            __builtin_amdgcn_global_load_async_to_lds_b128(
                (AS1 v4i*)(xsrc + t), (AS3 v4i*)(xdst + t), 0, 0);
        wait_asynccnt0();
#else
        for (int t = tid; t < (TB * XW) / 4; t += THREADS) xdst[t] = xsrc[t];
#endif
    } else {
        for (int t = tid; t < rows * XW; t += THREADS) xs[t] = x[(size_t)b0 * XW + t];
        for (int t = rows * XW + tid; t < TB * XW; t += THREADS) xs[t] = 0.f;
    }
    // ---- stage weights / biases / mask / LN params ----
    for (int t = tid; t < 4 * 81; t += THREADS) {
        wts[t] = Wq[t]; wts[324 + t] = Wk[t]; wts[648 + t] = Wv[t];
    }
    for (int t = tid; t < 4 * KVD; t += THREADS) {
        bis[t] = bq[t]; bis[36 + t] = bk[t]; bis[72 + t] = bv[t];
    }
    if (tid < NQ)  msk[tid] = mask[tid] * (-1e9f);
    if (tid < KVD) { gam[tid] = gamma[tid]; bet[tid] = beta[tid]; }
    if (tid == 0 && b0 + TB < B)
        __builtin_prefetch(x + ((size_t)b0 + TB) * XW, 0, 0);  // global_prefetch
    __syncthreads();

    // ---- phase 1: grouped projections via V_WMMA_F32_16X16X4_F32 ----
    // A[m,k]  lane layout: m = lane&15, k-pair = 2*(lane>>4) within each K-4 chunk
    // B[k,e]  lane layout: e = lane&15, same k-pair split
    // C/D[m,e]: VGPR r, lane l -> m = r + 8*(l>>4), e = l&15
    const int mrow  = lane & 15;
    const int khalf = (lane >> 4) << 1;       // 0 or 2
    for (int n = wave; n < NQ; n += NWAVES) {
        const int g = (n < 3) ? 0 : (n < 13) ? 1 : (n < 23) ? 2 : 3;
        v2f a[3];
        #pragma unroll
        for (int jk = 0; jk < 3; ++jk) {
            int k0 = 4 * jk + khalf;
            int k0c = (k0     < KVD) ? k0     : 0;
            int k1c = (k0 + 1 < KVD) ? k0 + 1 : 0;
            float a0 = xs[mrow * XW + n * KVD + k0c];
            float a1 = xs[mrow * XW + n * KVD + k1c];
            if (k0     >= KVD) a0 = 0.f;
            if (k0 + 1 >= KVD) a1 = 0.f;
            a[jk].x = a0; a[jk].y = a1;
        }
        #pragma unroll
        for (int mat = 0; mat < 3; ++mat) {
            const float* w  = &wts[mat * 324 + g * 81];
            const float* bb = &bis[mat * 36  + g * KVD];
            v2f bf[3];
            #pragma unroll
            for (int jk = 0; jk < 3; ++jk) {
                int k0 = 4 * jk + khalf;
                int ec  = (mrow < KVD) ? mrow : 0;
                int k0c = (k0     < KVD) ? k0     : 0;
                int k1c = (k0 + 1 < KVD) ? k0 + 1 : 0;
                float b0v = w[ec * KVD + k0c];        // B[k,e] = W[e][k]
                float b1v = w[ec * KVD + k1c];
                if (k0     >= KVD || mrow >= KVD) b0v = 0.f;
                if (k0 + 1 >= KVD || mrow >= KVD) b1v = 0.f;
                bf[jk].x = b0v; bf[jk].y = b1v;
            }
            float cinit = (mrow < KVD) ? bb[mrow] : 0.f;   // C[m,e] = bias[e]
            v8f c;
            #pragma unroll
            for (int r = 0; r < 8; ++r) c[r] = cinit;
            c = __builtin_amdgcn_wmma_f32_16x16x4_f32(false, a[0], false, bf[0], (short)0, c, false, false);
            c = __builtin_amdgcn_wmma_f32_16x16x4_f32(false, a[1], false, bf[1], (short)0, c, false, false);
            c = __builtin_amdgcn_wmma_f32_16x16x4_f32(false, a[2], false, bf[2], (short)0, c, false, false);
            float* dst = (mat == 0) ? qs : (mat == 1) ? ks_ : vs;
            if (mrow < KVD) {
                #pragma unroll
                for (int r = 0; r < 8; ++r) {
                    int m = r + ((lane >> 4) << 3);
                    dst[(n * TB + m) * KVD + mrow] = c[r];
                }
            }
        }
    }
    __syncthreads();

    // ---- phase 2: scores + softmax + attn@V + residual + LayerNorm ----
    const float scale = 1.0f / 3.0f;          // 1/sqrt(9)
    for (int mi = 0; mi < TB / NWAVES; ++mi) {
        const int m = wave * (TB / NWAVES) + mi;
        if (lane < NQ) {
            const int i = lane;
            float q[KVD];
            #pragma unroll
            for (int d = 0; d < KVD; ++d) q[d] = qs[(i * TB + m) * KVD + d] * scale;
            float s[NQ];
            float smax = -3.4e38f;
            #pragma unroll
            for (int j = 0; j < NQ; ++j) {
                float acc = msk[j];
                #pragma unroll
                for (int d = 0; d < KVD; ++d) acc += q[d] * ks_[(j * TB + m) * KVD + d];
                s[j] = acc;
                smax = fmaxf(smax, acc);
            }
            float ssum = 0.f;
            #pragma unroll
            for (int j = 0; j < NQ; ++j) { s[j] = __expf(s[j] - smax); ssum += s[j]; }
            const float inv = 1.f / ssum;
            float o[KVD] = {};
            #pragma unroll
            for (int j = 0; j < NQ; ++j) {
                float p = s[j];
                #pragma unroll
                for (int d = 0; d < KVD; ++d) o[d] += p * vs[(j * TB + m) * KVD + d];
            }
            float mu = 0.f;
            #pragma unroll
            for (int d = 0; d < KVD; ++d) {
                o[d] = o[d] * inv + xs[m * XW + i * KVD + d];   // + residual
                mu += o[d];
            }
            mu *= (1.f / KVD);
            float var = 0.f;
            #pragma unroll
            for (int d = 0; d < KVD; ++d) { float t = o[d] - mu; var += t * t; }
            const float rs = rsqrtf(var * (1.f / KVD) + 1e-5f);
            #pragma unroll
            for (int d = 0; d < KVD; ++d)
                xs[m * XW + i * KVD + d] = (o[d] - mu) * rs * gam[d] + bet[d];
        }
    }
    __syncthreads();

    // ---- output drain: async LDS->global if available, else coalesced stores ----
    if (rows == TB) {
        f4* odst = (f4*)(out + (size_t)b0 * XW);
        const f4* osrc = (const f4*)xs;
#if __has_builtin(__builtin_amdgcn_global_store_async_from_lds_b128)
        for (int t = tid; t < (TB * XW) / 4; t += THREADS)
            __builtin_amdgcn_global_store_async_from_lds_b128(
                (AS1 v4i*)(odst + t), (AS3 v4i*)(osrc + t), 0, 0);
        wait_asynccnt0();
#else
        for (int t = tid; t < (TB * XW) / 4; t += THREADS) odst[t] = osrc[t];
#endif
    } else {
        for (int t = tid; t < rows * XW; t += THREADS) out[(size_t)b0 * XW + t] = xs[t];
    }
}

extern "C" void kernel_launch(void* const* d_in, const int* in_sizes, int n_in,
                              void* d_out, int out_size, void* d_ws, size_t ws_size,
                              hipStream_t stream) {
    const float* x     = (const float*)d_in[0];
    const float* mask  = (const float*)d_in[1];
    const float* Wq    = (const float*)d_in[2];
    const float* bq    = (const float*)d_in[3];
    const float* Wk    = (const float*)d_in[4];
    const float* bk    = (const float*)d_in[5];
    const float* Wv    = (const float*)d_in[6];
    const float* bv    = (const float*)d_in[7];
    const float* gamma = (const float*)d_in[8];
    const float* beta  = (const float*)d_in[9];
    float* out = (float*)d_out;
    const int B = in_sizes[0] / XW;
    const int grid = (B + TB - 1) / TB;
    hipLaunchKernelGGL(attn_small_kernel, dim3(grid), dim3(THREADS), 0, stream,
                       x, mask, Wq, bq, Wk, bk, Wv, bv, gamma, beta, out, B);
}